// SpikingActivation_4947802325587
// MI455X (gfx1250) — compile-verified
//
#include <hip/hip_runtime.h>
#include <cstdint>
#include <cstddef>

// Problem constants (reference: B=32, T=2048, D=1024, fp32)
#define B_  32
#define T_  2048
#define D_  1024
#define DW  16                 // features per workgroup
#define NC  32                 // time chunks per sequence
#define CL  64                 // timesteps per chunk (NC*CL == T_)
#define NTHREADS (DW * NC)     // 512 threads = 16 waves (wave32)

// Single-pass chunked IIR scan, one workgroup per (batch b, 16-feature group).
//   Phase A: async-copy the 128KB tile into LDS (ASYNCcnt fire-and-forget)
//            LDS layout is chunk-minor [j][c][dl] -> conflict-free reads
//   Phase B: per-thread local scan of its 64-step chunk (zero init) -> aggregate
//   Phase C: serial carry combine across the 32 chunks (16 threads, s^64 step)
//   Phase D: per-thread rescan with true carry, non-temporal store to HBM
__global__ __launch_bounds__(NTHREADS)
void ema_scan_kernel(const float* __restrict__ x,
                     const float* __restrict__ lv,
                     const float* __restrict__ sv,
                     float* __restrict__ out) {
  // Tile layout: word(j, c, dl) = j*512 + c*16 + dl.
  // At a fixed scan step j, the 512 threads (32 lanes/wave) read 512
  // consecutive words -> every wave hits 32 distinct banks (no conflicts),
  // unlike [t][dl] where the two half-waves (delta-t = 64 rows) always
  // alias onto the same 16-bank group.
  __shared__ float xs[T_ * DW];    // 128 KB input tile
  __shared__ float carr[NC * DW];  //   2 KB chunk aggregates / carries

  const int dl    = threadIdx.x;           // 0..15 feature-in-group (lane-fast)
  const int c     = threadIdx.y;           // 0..31 chunk id
  const int tid   = dl + DW * c;           // 0..511
  const int b     = blockIdx.y;
  const int dbase = blockIdx.x * DW;
  const int d     = dbase + dl;
  const size_t row = (size_t)b * T_;

  // ---------------- Phase A: async tile load (global -> LDS) ----------------
  // 8192 16-byte chunks; chunk m decomposes as m = j*128 + c*4 + seg.
  // LDS destination is chunk m itself (identity); global source is row
  // t = c*64 + j, 16-byte segment seg of this WG's 64-byte feature slice.
  // Consecutive lanes cover contiguous 64B runs of each global row.
  {
    const unsigned xs_base = (unsigned)(size_t)(void*)xs;  // low 32b = LDS offset
    #pragma unroll
    for (int r = 0; r < 16; ++r) {
      const int m   = r * NTHREADS + tid;    // 0..8191
      const int j   = m >> 7;                // 0..63  scan step
      const int cc  = (m >> 2) & (NC - 1);   // 0..31  chunk
      const int seg = (m & 3) << 2;          // float offset 0,4,8,12
      const int t   = cc * CL + j;
      const float* gp = x + ((row + (size_t)t) * D_ + dbase + seg);
      const unsigned lo = xs_base + (unsigned)(m << 4);
      asm volatile("global_load_async_to_lds_b128 %0, %1, off"
                   :: "v"(lo), "v"(gp)
                   : "memory");
    }
  }
  asm volatile("s_wait_asynccnt 0" ::: "memory");  // this wave's copies landed
  __syncthreads();                                 // all waves' copies landed

  // Per-feature smoothing coefficient
  const float s = 1.0f / (1.0f + __expf(-sv[d]));
  const float a = 1.0f - s;
  float sp = s;                       // s^CL = s^64 via 6 squarings
  #pragma unroll
  for (int i = 0; i < 6; ++i) sp = sp * sp;

  // ---------------- Phase B: local chunk scan (zero initial level) ----------
  {
    float lvl = 0.0f;
    const float* xcol = xs + c * DW + dl;   // stride NC*DW = 512 words per j
    #pragma unroll 8
    for (int j = 0; j < CL; ++j) {
      lvl = fmaf(s, lvl, a * xcol[j * (NC * DW)]);
    }
    carr[c * DW + dl] = lvl;                // L_c : chunk-final with zero init
  }
  __syncthreads();

  // ---------------- Phase C: carry combine across chunks --------------------
  // F_{-1} = level_var[d]; F_c = L_c + s^CL * F_{c-1}; carry-in(c) = F_{c-1}.
  if (c == 0) {
    float F = lv[d];
    #pragma unroll
    for (int cc = 0; cc < NC; ++cc) {
      const float L = carr[cc * DW + dl];
      carr[cc * DW + dl] = F;               // carry-in for chunk cc
      F = fmaf(sp, F, L);
    }
  }
  __syncthreads();

  // ---------------- Phase D: rescan with true carry, stream out -------------
  {
    float lvl = carr[c * DW + dl];
    const float* xcol = xs + c * DW + dl;
    float* orow = out + (row + (size_t)(c * CL)) * D_ + d;
    #pragma unroll 8
    for (int j = 0; j < CL; ++j) {
      lvl = fmaf(s, lvl, a * xcol[j * (NC * DW)]);
      __builtin_nontemporal_store(lvl, orow + (size_t)j * D_);  // write-once stream
    }
  }
}

extern "C" void kernel_launch(void* const* d_in, const int* in_sizes, int n_in,
                              void* d_out, int out_size, void* d_ws, size_t ws_size,
                              hipStream_t stream) {
  const float* x  = (const float*)d_in[0];  // inputs        [B,T,D] fp32
  const float* lv = (const float*)d_in[1];  // level_var     [D]     fp32
  const float* sv = (const float*)d_in[2];  // smoothing_var [D]     fp32
  float* out = (float*)d_out;               // [B,T,D] fp32

  dim3 grid(D_ / DW, B_, 1);   // 64 x 32 = 2048 workgroups
  dim3 block(DW, NC, 1);       // 512 threads = 16 waves
  ema_scan_kernel<<<grid, block, 0, stream>>>(x, lv, sv, out);
}